// Sylo_75565654606093
// MI455X (gfx1250) — compile-verified
//
#include <hip/hip_runtime.h>

typedef __attribute__((ext_vector_type(2))) float v2f;
typedef __attribute__((ext_vector_type(8))) float v8f;

#define N_    16
#define CIN   4
#define COUT  32
#define H_    400
#define RANK  4
#define TILE  16
#define NT    (H_ / TILE)   // 25
#define TILES (NT * NT)     // 625
#define WPB   8             // wave-tiles per 256-thread block

__device__ __forceinline__ v8f wmma_t16(v2f a, v2f b) {
  v8f z = {0.f, 0.f, 0.f, 0.f, 0.f, 0.f, 0.f, 0.f};
  // D(16x16) = A(16x4) x B(4x16) + 0 : one instruction per rank-4 T tile
  return __builtin_amdgcn_wmma_f32_16x16x4_f32(false, a, false, b,
                                               (short)0, z, false, false);
}

// Pass A: accumulate rows[n,o,i] += sum_j diag ; cols[n,o,j] += sum_i diag
__global__ __launch_bounds__(256) void sylo_rowcol_kernel(
    const float* __restrict__ x, const float* __restrict__ wL,
    float* __restrict__ rows, float* __restrict__ cols) {
  const int wave = threadIdx.x >> 5;
  const int lane = threadIdx.x & 31;
  const int tile = blockIdx.x * WPB + wave;
  if (tile >= TILES) return;
  const int n  = blockIdx.y;
  const int i0 = (tile / NT) * TILE;
  const int j0 = (tile % NT) * TILE;
  const int hi = lane >> 4;     // which 16-lane half
  const int lm = lane & 15;
  const int j  = j0 + lm;
  const int k0 = 2 * hi;        // K pair held by this half (A and B layouts)

  // X tile: lane holds rows i = i0 + v + 8*hi, column j, per channel
  float xv[CIN][8];
#pragma unroll
  for (int c = 0; c < CIN; ++c) {
    const float* xp = x + ((n * CIN + c) * H_ + i0 + 8 * hi) * H_ + j;
#pragma unroll
    for (int v = 0; v < 8; ++v) xv[c][v] = xp[v * H_];
  }

  const bool b1 = lane & 1;
  const bool b2 = lane & 2;
  const bool b4 = lane & 4;

  for (int o = 0; o < COUT; ++o) {
    v8f e = {0.f, 0.f, 0.f, 0.f, 0.f, 0.f, 0.f, 0.f};
#pragma unroll
    for (int c = 0; c < CIN; ++c) {
      const float* lp = wL + (o * CIN + c) * H_ * RANK;
      v2f a = *(const v2f*)(lp + (i0 + lm) * RANK + k0);  // Li rows
      v2f b = *(const v2f*)(lp + (j0 + lm) * RANK + k0);  // Lj rows
      v8f d = wmma_t16(a, b);                             // T tile
#pragma unroll
      for (int v = 0; v < 8; ++v) e[v] += xv[c][v] * d[v]; // diag tile
    }

    // ---- column sums: lane-local over v, then combine halves ----
    float s = e[0] + e[1] + e[2] + e[3] + e[4] + e[5] + e[6] + e[7];
    s += __shfl_xor(s, 16, 32);

    // ---- row sums: fused register/lane butterfly (8 shuffles total) ----
    // stage 1 (xor 1): 8 regs -> 4
    float g0, f0, f1, f2, f3;
    f0 = b1 ? e[1] : e[0]; g0 = b1 ? e[0] : e[1]; f0 += __shfl_xor(g0, 1, 32);
    f1 = b1 ? e[3] : e[2]; g0 = b1 ? e[2] : e[3]; f1 += __shfl_xor(g0, 1, 32);
    f2 = b1 ? e[5] : e[4]; g0 = b1 ? e[4] : e[5]; f2 += __shfl_xor(g0, 1, 32);
    f3 = b1 ? e[7] : e[6]; g0 = b1 ? e[6] : e[7]; f3 += __shfl_xor(g0, 1, 32);
    // stage 2 (xor 2): 4 -> 2
    float h0, h1;
    h0 = b2 ? f1 : f0; g0 = b2 ? f0 : f1; h0 += __shfl_xor(g0, 2, 32);
    h1 = b2 ? f3 : f2; g0 = b2 ? f2 : f3; h1 += __shfl_xor(g0, 2, 32);
    // stage 3 (xor 4): 2 -> 1
    float r;
    r = b4 ? h1 : h0; g0 = b4 ? h0 : h1; r += __shfl_xor(g0, 4, 32);
    // stage 4 (xor 8): plain reduce; lane l now holds rowsum[(l&7) + 8*hi]
    r += __shfl_xor(r, 8, 32);

    const int base = (n * COUT + o) * H_;
    if (lm < 8) atomicAdd(&rows[base + i0 + (lm & 7) + 8 * hi], r);
    if (hi == 0) atomicAdd(&cols[base + j], s);
  }
}

// Pass B: out = rows[i] + cols[j] - diag   (bias pre-folded into rows/cols)
__global__ __launch_bounds__(256) void sylo_out_kernel(
    const float* __restrict__ x, const float* __restrict__ wL,
    const float* __restrict__ rows, const float* __restrict__ cols,
    float* __restrict__ out) {
  const int wave = threadIdx.x >> 5;
  const int lane = threadIdx.x & 31;
  const int tile = blockIdx.x * WPB + wave;
  if (tile >= TILES) return;
  const int n  = blockIdx.y;
  const int i0 = (tile / NT) * TILE;
  const int j0 = (tile % NT) * TILE;
  const int hi = lane >> 4;
  const int lm = lane & 15;
  const int j  = j0 + lm;
  const int k0 = 2 * hi;

  float xv[CIN][8];
#pragma unroll
  for (int c = 0; c < CIN; ++c) {
    const float* xp = x + ((n * CIN + c) * H_ + i0 + 8 * hi) * H_ + j;
#pragma unroll
    for (int v = 0; v < 8; ++v) xv[c][v] = xp[v * H_];
  }

  for (int o = 0; o < COUT; ++o) {
    v8f e = {0.f, 0.f, 0.f, 0.f, 0.f, 0.f, 0.f, 0.f};
#pragma unroll
    for (int c = 0; c < CIN; ++c) {
      const float* lp = wL + (o * CIN + c) * H_ * RANK;
      v2f a = *(const v2f*)(lp + (i0 + lm) * RANK + k0);
      v2f b = *(const v2f*)(lp + (j0 + lm) * RANK + k0);
      v8f d = wmma_t16(a, b);
#pragma unroll
      for (int v = 0; v < 8; ++v) e[v] += xv[c][v] * d[v];
    }
    const int base = (n * COUT + o) * H_;
    const float cv = cols[base + j];                 // includes bias/2
    float* op = out + (size_t)((n * COUT + o) * H_ + i0 + 8 * hi) * H_ + j;
#pragma unroll
    for (int v = 0; v < 8; ++v) {
      op[v * H_] = (rows[base + i0 + 8 * hi + v] + cv) - e[v];
    }
  }
}

// Init rows/cols to bias[o]/2 so pass B needs no separate bias load.
__global__ void sylo_init_kernel(const float* __restrict__ bias,
                                 float* __restrict__ rows,
                                 float* __restrict__ cols) {
  int idx = blockIdx.x * blockDim.x + threadIdx.x;
  if (idx < N_ * COUT * H_) {
    int o = (idx / H_) % COUT;
    float b = 0.5f * bias[o];
    rows[idx] = b;
    cols[idx] = b;
  }
}

extern "C" void kernel_launch(void* const* d_in, const int* in_sizes, int n_in,
                              void* d_out, int out_size, void* d_ws, size_t ws_size,
                              hipStream_t stream) {
  (void)in_sizes; (void)n_in; (void)out_size; (void)ws_size;
  const float* x    = (const float*)d_in[0];   // (16,4,400,400) f32
  const float* wL   = (const float*)d_in[1];   // (32,4,400,4)   f32
  const float* bias = (const float*)d_in[2];   // (32,)          f32
  float* out  = (float*)d_out;                 // (16,32,400,400) f32
  float* rows = (float*)d_ws;                  // N*COUT*H
  float* cols = rows + N_ * COUT * H_;         // N*COUT*H  (total 1.6 MB)

  const int in_elems = N_ * COUT * H_;
  sylo_init_kernel<<<(in_elems + 255) / 256, 256, 0, stream>>>(bias, rows, cols);

  dim3 grid((TILES + WPB - 1) / WPB, N_);
  sylo_rowcol_kernel<<<grid, 256, 0, stream>>>(x, wL, rows, cols);
  sylo_out_kernel<<<grid, 256, 0, stream>>>(x, wL, rows, cols, out);
}